// PaletteExtractor_14482629722925
// MI455X (gfx1250) — compile-verified
//
#include <hip/hip_runtime.h>
#include <stdint.h>

// Problem constants (match reference: B=512, H=W=128, C=4)
#define BATCH   512
#define PIX     16384                 // H*W
#define NTHR    1024                  // 32 waves of 32
#define NCHUNK  (PIX / NTHR)          // 16
#define NSLOTS  24576                 // hash slots, load factor <= 0.67 worst case
#define NWORDS  (PIX / 32)            // 512 flag words

static __device__ constexpr unsigned long long EMPTY64 = ~0ull;

__device__ __forceinline__ unsigned hash32(unsigned x) {
  x *= 0x9E3779B1u;
  x ^= x >> 16;
  x *= 0x85EBCA77u;
  x ^= x >> 13;
  return x;
}

// gfx1250 async copy global -> LDS (16B per lane), tracked by ASYNCcnt.
__device__ __forceinline__ void async_ld16(void* lds, const void* g) {
  unsigned l = (unsigned)(uintptr_t)lds;                 // low 32 bits of generic LDS ptr = LDS byte addr
  unsigned long long ga = (unsigned long long)(uintptr_t)g;
  asm volatile("global_load_async_to_lds_b128 %0, %1, off"
               :: "v"(l), "v"(ga)
               : "memory");
}
#define ASYNC_WAIT(n) asm volatile("s_wait_asynccnt " #n ::: "memory")

__global__ __launch_bounds__(NTHR, 1) void palette_kernel(const float* __restrict__ in,
                                                          float* __restrict__ out) {
  // ~292 KB of the 320 KB WGP LDS
  __shared__ unsigned long long hashTab[NSLOTS];   // 192 KB : packed (key<<14 | minPixelIdx)
  __shared__ unsigned keysBuf[PIX];                //  64 KB : per-pixel 32-bit color key
  __shared__ float4 stage[2][NTHR];                //  32 KB : double-buffered async stage
  __shared__ unsigned flagWords[NWORDS];           //   2 KB : first-occurrence bitmask
  __shared__ unsigned scanA[NWORDS];               //   2 KB : inclusive popcount scan

  const int t = threadIdx.x;
  const int b = blockIdx.x;
  const float4* img = (const float4*)(in) + (size_t)b * PIX;
  float4* outImg = (float4*)(out) + (size_t)b * PIX;
  float* outCnt = out + (size_t)BATCH * PIX * 4;

  // Kick off chunk 0 DMA; overlap with hash-table clear.
  async_ld16(&stage[0][t], img + t);

  for (int i = t; i < NSLOTS; i += NTHR) hashTab[i] = EMPTY64;
  __syncthreads();

  // ---- Phase A: stream pixels via async LDS stage; build keys + (key -> min idx) hash ----
  for (int c = 0; c < NCHUNK; ++c) {
    if (c + 1 < NCHUNK) {
      async_ld16(&stage[(c + 1) & 1][t], img + (c + 1) * NTHR + t);
      ASYNC_WAIT(1);          // chunk c complete, chunk c+1 in flight
    } else {
      ASYNC_WAIT(0);
    }
    float4 px = stage[c & 1][t];
    const int p = c * NTHR + t;
    // tf.cast truncation toward zero; (x+1)*127.5 >= 0 so plain int cast matches.
    unsigned c0 = (unsigned)(int)((px.x + 1.0f) * 127.5f);
    unsigned c1 = (unsigned)(int)((px.y + 1.0f) * 127.5f);
    unsigned c2 = (unsigned)(int)((px.z + 1.0f) * 127.5f);
    unsigned c3 = (unsigned)(int)((px.w + 1.0f) * 127.5f);
    unsigned key = (c0 << 24) | (c1 << 16) | (c2 << 8) | c3;   // 256^3..256^0 weights
    keysBuf[p] = key;

    unsigned long long packed = ((unsigned long long)key << 14) | (unsigned)p; // idx fits 14 bits
    unsigned h = hash32(key) % NSLOTS;
    for (;;) {
      unsigned long long cur = hashTab[h];
      if (cur == EMPTY64) {
        unsigned long long old = atomicCAS(&hashTab[h], EMPTY64, packed);  // ds_cmpstore_rtn_b64
        if (old == EMPTY64) break;
        cur = old;
      }
      if ((cur >> 14) == (unsigned long long)key) {
        atomicMin(&hashTab[h], packed);                                    // ds_min_u64 on idx bits
        break;
      }
      ++h; if (h == NSLOTS) h = 0;
    }
  }
  __syncthreads();

  // ---- Phase B: flag first occurrences, pack 32 flags/word via wave32 ballot ----
  for (int c = 0; c < NCHUNK; ++c) {
    const int p = c * NTHR + t;
    unsigned key = keysBuf[p];
    unsigned h = hash32(key) % NSLOTS;
    unsigned minIdx;
    for (;;) {
      unsigned long long cur = hashTab[h];
      if ((cur >> 14) == (unsigned long long)key) { minIdx = (unsigned)(cur & 0x3FFFull); break; }
      ++h; if (h == NSLOTS) h = 0;
    }
    unsigned long long bal = __ballot(minIdx == (unsigned)p);
    if ((t & 31) == 0) flagWords[p >> 5] = (unsigned)bal;
  }
  __syncthreads();

  // ---- Phase C: inclusive scan of per-word popcounts (512 words, Hillis-Steele) ----
  if (t < NWORDS) scanA[t] = (unsigned)__popc(flagWords[t]);
  __syncthreads();
  for (int off = 1; off < NWORDS; off <<= 1) {
    unsigned v = 0;
    if (t < NWORDS) {
      v = scanA[t];
      if (t >= off) v += scanA[t - off];
    }
    __syncthreads();
    if (t < NWORDS) scanA[t] = v;
    __syncthreads();
  }
  const int count = (int)scanA[NWORDS - 1];

  // ---- Phase D: scatter unique colors by first-appearance rank; zero the tail ----
  for (int c = 0; c < NCHUNK; ++c) {
    const int p = c * NTHR + t;
    const int w = p >> 5;
    unsigned m = flagWords[w];
    if ((m >> (p & 31)) & 1u) {
      int rank = (int)(scanA[w] - (unsigned)__popc(m))            // exclusive word prefix
               + __popc(m & ((1u << (p & 31)) - 1u));             // rank within word
      unsigned key = keysBuf[p];
      float4 col;
      col.x = (float)((key >> 24) & 255u) * (1.0f / 127.5f) - 1.0f;
      col.y = (float)((key >> 16) & 255u) * (1.0f / 127.5f) - 1.0f;
      col.z = (float)((key >>  8) & 255u) * (1.0f / 127.5f) - 1.0f;
      col.w = (float)( key        & 255u) * (1.0f / 127.5f) - 1.0f;
      outImg[rank] = col;
    }
  }
  // rows [count, PIX) are zero padding; disjoint from scatter range -> no race
  for (int r = count + t; r < PIX; r += NTHR) {
    outImg[r] = make_float4(0.0f, 0.0f, 0.0f, 0.0f);
  }
  if (t == 0) outCnt[b] = (float)count;
}

extern "C" void kernel_launch(void* const* d_in, const int* in_sizes, int n_in,
                              void* d_out, int out_size, void* d_ws, size_t ws_size,
                              hipStream_t stream) {
  (void)in_sizes; (void)n_in; (void)d_ws; (void)ws_size; (void)out_size;
  const float* in = (const float*)d_in[0];
  float* out = (float*)d_out;
  hipLaunchKernelGGL(palette_kernel, dim3(BATCH), dim3(NTHR), 0, stream, in, out);
}